// ConvWithFFT_37546604101897
// MI455X (gfx1250) — compile-verified
//
#include <hip/hip_runtime.h>
#include <hip/hip_bf16.h>

// ------------------------------------------------------------------
// FFT-based linear convolution on MI455X (gfx1250).
//   y = ifft( fft(pad(a)) * fft(pad(x)) ).real[:n]
// n = 2^23, padded length NF = 2^24 = 4096 x 4096.
//
// Bandwidth-bound: ~2.4 GB of HBM traffic total (~100 us at 23.3 TB/s).
// z = a + i*x packing -> ONE forward FFT + ONE inverse FFT.
// 4096-pt row FFTs in LDS, 3 fully-unrolled Stockham radix-16 stages,
// each a complex 16x16 DFT-matrix product via v_wmma_f32_16x16x4_f32.
// Row input is streamed global->LDS with GLOBAL_LOAD_ASYNC_TO_LDS_B32
// (ASYNCcnt) when the toolchain exposes the builtin.
// ------------------------------------------------------------------

#define NSIG  (1 << 23)   // real signal length
#define NF    (1 << 24)   // padded complex FFT length
#define DIM   4096        // NF = DIM x DIM
#define PI2f  6.28318530717958647692f

typedef float v8f __attribute__((ext_vector_type(8)));
typedef float v2f __attribute__((ext_vector_type(2)));

#if defined(__has_builtin)
#if __has_builtin(__builtin_amdgcn_global_load_async_to_lds_b32) && \
    __has_builtin(__builtin_amdgcn_s_wait_asynccnt)
#define HAVE_ASYNC_LDS 1
#endif
#endif

// D = A(16x4) * B(4x16) + C   -- fp32 WMMA, K accumulated in chunks of 4.
__device__ __forceinline__ v8f wmma4(v2f a, v2f b, v8f c) {
  // 8-arg pattern: (neg_a, A, neg_b, B, c_mod, C, reuse_a, reuse_b)
  return __builtin_amdgcn_wmma_f32_16x16x4_f32(
      false, a, false, b, (short)0, c, false, false);
}

// ------------------------------------------------------------------
// pack: z[i] = a[i] + i*x[i] for i < NSIG, else 0
// ------------------------------------------------------------------
__global__ __launch_bounds__(256) void pack_kernel(
    const float* __restrict__ a, const float* __restrict__ x,
    float2* __restrict__ z) {
  int i = blockIdx.x * 256 + threadIdx.x;
  float2 v;
  if (i < NSIG) { v.x = a[i]; v.y = x[i]; } else { v.x = 0.0f; v.y = 0.0f; }
  z[i] = v;
}

// ------------------------------------------------------------------
// tiled 4096x4096 complex transpose (coalesced both sides)
// ------------------------------------------------------------------
__global__ __launch_bounds__(256) void transpose_kernel(
    const float2* __restrict__ in, float2* __restrict__ out) {
  __shared__ float2 tile[32][33];
  const int bx = blockIdx.x * 32, by = blockIdx.y * 32;
  const int tx = threadIdx.x & 31, ty = threadIdx.x >> 5;
  for (int j = ty; j < 32; j += 8)
    tile[j][tx] = in[(size_t)(by + j) * DIM + bx + tx];
  __syncthreads();
  for (int j = ty; j < 32; j += 8)
    out[(size_t)(bx + j) * DIM + by + tx] = tile[tx][j];
}

// ------------------------------------------------------------------
// One Stockham radix-16 stage: 16 complex 16x16 DFT matmuls across the
// 256 columns (2 matrices per wave32).  Reads base + 256*s; the output
// index / twiddle are stage-specific compile-time variants.
// ------------------------------------------------------------------
template<int ST>
__device__ __forceinline__ void fft_stage(
    const float* __restrict__ reIn, const float* __restrict__ imIn,
    float* __restrict__ reOut, float* __restrict__ imOut,
    const v2f* __restrict__ Ar, const v2f* __restrict__ Ai,
    const v2f* __restrict__ Ain,
    int wave, int hi, int c, float dir) {
#pragma unroll
  for (int mm = 0; mm < 2; ++mm) {
    const int m    = wave * 2 + mm;
    const int base = m * 16 + c;

    // B(4x16 chunk) layout: VGPR j -> K = 4*kk + 2*hi + j, N = lane&15
    v2f Br[4], Bi[4];
#pragma unroll
    for (int kk = 0; kk < 4; ++kk) {
      int i0 = base + 256 * (kk * 4 + 2 * hi);
      Br[kk] = (v2f){reIn[i0], reIn[i0 + 256]};
      Bi[kk] = (v2f){imIn[i0], imIn[i0 + 256]};
    }

    v8f Cr = {0,0,0,0,0,0,0,0};
    v8f Ci = {0,0,0,0,0,0,0,0};
    // Cr = Dr*Xr - Di*Xi ; Ci = Dr*Xi + Di*Xr   (16 WMMAs)
#pragma unroll
    for (int kk = 0; kk < 4; ++kk) Cr = wmma4(Ar[kk],  Br[kk], Cr);
#pragma unroll
    for (int kk = 0; kk < 4; ++kk) Cr = wmma4(Ain[kk], Bi[kk], Cr);
#pragma unroll
    for (int kk = 0; kk < 4; ++kk) Ci = wmma4(Ar[kk],  Bi[kk], Ci);
#pragma unroll
    for (int kk = 0; kk < 4; ++kk) Ci = wmma4(Ai[kk],  Br[kk], Ci);

    // C(16x16) layout: VGPR r -> M = r + 8*hi, N = lane&15
#pragma unroll
    for (int r = 0; r < 8; ++r) {
      const int k1 = r + 8 * hi;
      float cr = Cr[r], ci = Ci[r];
      int outIdx;
      if constexpr (ST == 0) {        // twiddle W_4096^{base*k1}
        outIdx = base * 16 + k1;
        float ang = dir * (PI2f / 4096.0f) * (float)((base * k1) & 4095);
        float s, cs;
        __sincosf(ang, &s, &cs);
        float t = cr * cs - ci * s;
        ci = cr * s + ci * cs;
        cr = t;
      } else if constexpr (ST == 1) { // twiddle W_256^{(base>>4)*k1}
        outIdx = (base >> 4) * 256 + k1 * 16 + (base & 15);
        float ang = dir * (PI2f / 256.0f) * (float)((base >> 4) * k1);
        float s, cs;
        __sincosf(ang, &s, &cs);
        float t = cr * cs - ci * s;
        ci = cr * s + ci * cs;
        cr = t;
      } else {                        // final stage: no twiddle
        outIdx = k1 * 256 + base;
      }
      reOut[outIdx] = cr;
      imOut[outIdx] = ci;
    }
  }
}

// ------------------------------------------------------------------
// 4096-point in-place row FFT (natural order, Stockham radix-16 x3)
// dir = -1: forward (W = e^{-2pi i/N}),  dir = +1: inverse (unscaled)
// GTW: multiply output element k by exp(dir * 2pi i * row * k / NF)
// ------------------------------------------------------------------
template<int GTW>
__global__ __launch_bounds__(256) void rowfft_kernel(
    float2* __restrict__ data, float dir) {
  __shared__ float lds[4 * 4096];   // 64 KB: re/im ping + re/im pong
  float* reA = lds;
  float* imA = lds + 4096;
  float* reB = lds + 8192;
  float* imB = lds + 12288;

  const int tid  = threadIdx.x;
  const int lane = tid & 31;
  const int wave = tid >> 5;
  const int hi   = lane >> 4;   // half-wave select (K split)
  const int c    = lane & 15;   // column / M index within fragment
  const int row  = blockIdx.x;
  float2* rowp = data + (size_t)row * DIM;

  // ---- stream row into LDS, deinterleaving re/im ----
#if HAVE_ASYNC_LDS
  {
    // gfx1250 async copy: global -> LDS without VGPR staging (ASYNCcnt)
    typedef __attribute__((address_space(1))) int GI;
    typedef __attribute__((address_space(3))) int LI;
    const float* g = (const float*)rowp;
#pragma unroll
    for (int j = 0; j < 16; ++j) {
      int idx = tid + 256 * j;
      __builtin_amdgcn_global_load_async_to_lds_b32(
          (GI*)(g + 2 * idx),     (LI*)(reA + idx), 0, 0);
      __builtin_amdgcn_global_load_async_to_lds_b32(
          (GI*)(g + 2 * idx + 1), (LI*)(imA + idx), 0, 0);
    }
    __builtin_amdgcn_s_wait_asynccnt(0);
  }
#else
#pragma unroll
  for (int j = 0; j < 16; ++j) {
    int idx = tid + 256 * j;
    float2 v = rowp[idx];
    reA[idx] = v.x;
    imA[idx] = v.y;
  }
#endif

  // ---- DFT16 matrix as A-operand fragments: D[M][K] = e^{dir*2pi i M K/16}
  // A(16x4) layout: lanes 0-15 M=0..15, VGPR j -> K = 4*kk + 2*hi + j
  v2f Ar[4], Ai[4], Ain[4];
  {
    const int M = c;
#pragma unroll
    for (int kk = 0; kk < 4; ++kk) {
      int K0 = kk * 4 + 2 * hi;
      float s0, c0, s1, c1;
      float a0 = dir * (PI2f / 16.0f) * (float)((M * K0) & 15);
      float a1 = dir * (PI2f / 16.0f) * (float)((M * (K0 + 1)) & 15);
      __sincosf(a0, &s0, &c0);
      __sincosf(a1, &s1, &c1);
      Ar[kk]  = (v2f){ c0,  c1};
      Ai[kk]  = (v2f){ s0,  s1};
      Ain[kk] = (v2f){-s0, -s1};
    }
  }

  // ---- 3 fully-unrolled Stockham radix-16 stages (ping-pong in LDS) ----
  __syncthreads();
  fft_stage<0>(reA, imA, reB, imB, Ar, Ai, Ain, wave, hi, c, dir);
  __syncthreads();
  fft_stage<1>(reB, imB, reA, imA, Ar, Ai, Ain, wave, hi, c, dir);
  __syncthreads();
  fft_stage<2>(reA, imA, reB, imB, Ar, Ai, Ain, wave, hi, c, dir);
  __syncthreads();

  // ---- optional four-step global twiddle, then store (coalesced) ----
  const float gscale = dir * (PI2f / 16777216.0f);
#pragma unroll
  for (int j = 0; j < 16; ++j) {
    int k = tid + 256 * j;
    float re = reB[k], im = imB[k];
    if constexpr (GTW != 0) {
      int kr = (row * k) & (NF - 1);     // exact integer phase reduction
      float ang = gscale * (float)kr;
      float s, cs;
      __sincosf(ang, &s, &cs);
      float r2 = re * cs - im * s;
      im = re * s + im * cs;
      re = r2;
    }
    rowp[k] = make_float2(re, im);
  }
}

// ------------------------------------------------------------------
// Hermitian split + pointwise product (+ fold 1/N for the inverse).
// Spectrum is stored transposed: position p = k1*4096 + k2 holds true
// frequency k = k2*4096 + k1.  Partner of k is (NF - k) mod NF.
// ------------------------------------------------------------------
__global__ __launch_bounds__(256) void combine_kernel(
    const float2* __restrict__ spec, float2* __restrict__ outspec) {
  const int p  = blockIdx.x * 256 + threadIdx.x;
  const int k1 = p >> 12, k2 = p & 4095;
  const int k  = (k2 << 12) | k1;
  const int km = (NF - k) & (NF - 1);
  const int pp = ((km & 4095) << 12) | (km >> 12);
  const float2 Z  = spec[p];
  const float2 Zr = spec[pp];
  // Fa = (Z + conj(Zr))/2 ; Fx = (Z - conj(Zr))/(2i)
  const float far = 0.5f * (Z.x + Zr.x);
  const float fai = 0.5f * (Z.y - Zr.y);
  const float fxr = 0.5f * (Z.y + Zr.y);
  const float fxi = -0.5f * (Z.x - Zr.x);
  const float invN = 1.0f / 16777216.0f;
  outspec[p] = make_float2((far * fxr - fai * fxi) * invN,
                           (far * fxi + fai * fxr) * invN);
}

// ------------------------------------------------------------------
// extract: out[n1*4096 + n2] = Z[n2][n1].re  for n1 < 2048 (first n samples)
// tiled transpose so both global streams stay coalesced
// ------------------------------------------------------------------
__global__ __launch_bounds__(256) void extract_kernel(
    const float2* __restrict__ z, float* __restrict__ out) {
  __shared__ float tile[32][33];
  const int n1b = blockIdx.x * 32;   // n1 in [0, 2048)
  const int n2b = blockIdx.y * 32;   // n2 in [0, 4096)
  const int tx = threadIdx.x & 31, ty = threadIdx.x >> 5;
  for (int j = ty; j < 32; j += 8)
    tile[j][tx] = z[(size_t)(n2b + j) * DIM + n1b + tx].x;
  __syncthreads();
  for (int j = ty; j < 32; j += 8)
    out[(size_t)(n1b + j) * DIM + n2b + tx] = tile[tx][j];
}

// ------------------------------------------------------------------
extern "C" void kernel_launch(void* const* d_in, const int* in_sizes, int n_in,
                              void* d_out, int out_size, void* d_ws, size_t ws_size,
                              hipStream_t stream) {
  const float* a = (const float*)d_in[0];
  const float* x = (const float*)d_in[1];
  float* out = (float*)d_out;

  float2* Z0 = (float2*)d_ws;   // 128 MB
  float2* Z1 = Z0 + NF;         // 128 MB

  // forward FFT of packed z = a + i*x  (four-step: col pass, twiddle, row pass)
  pack_kernel      <<<NF / 256, 256, 0, stream>>>(a, x, Z0);
  transpose_kernel <<<dim3(128, 128), 256, 0, stream>>>(Z0, Z1);
  rowfft_kernel<1> <<<DIM, 256, 0, stream>>>(Z1, -1.0f);
  transpose_kernel <<<dim3(128, 128), 256, 0, stream>>>(Z1, Z0);
  rowfft_kernel<0> <<<DIM, 256, 0, stream>>>(Z0, -1.0f);

  // Hermitian split -> Fa, Fx -> pointwise product (1/N folded in)
  combine_kernel   <<<NF / 256, 256, 0, stream>>>(Z0, Z1);

  // inverse FFT (same structure, conjugated twiddles, steps mirrored)
  rowfft_kernel<1> <<<DIM, 256, 0, stream>>>(Z1, +1.0f);
  transpose_kernel <<<dim3(128, 128), 256, 0, stream>>>(Z1, Z0);
  rowfft_kernel<0> <<<DIM, 256, 0, stream>>>(Z0, +1.0f);

  // real part of first n samples (transposed layout -> natural order)
  extract_kernel   <<<dim3(64, 128), 256, 0, stream>>>(Z0, out);
}